// MultiHeadAttention_19499151524021
// MI455X (gfx1250) — compile-verified
//
#include <hip/hip_runtime.h>
#include <math.h>

// ---------------------------------------------------------------------------
// GAT multi-head attention for MI455X (gfx1250, wave32).
// N=50000 nodes, E=1.6M edges, D_X=128, D_H=64, K=4 heads.
//
// Roofline: edge gather/scatter (~3.3GB) dominates, but Wx (51MB) and agg
// (51MB) fit in the 192MB L2 -> edge phase runs at L2 bandwidth. The two GEMMs
// (5 GFLOP total) use V_WMMA_F32_16X16X4_F32 to keep full fp32 precision.
//
// Phases:
//   1. gemm_xw      : Wx[N,256] = x[N,128] @ Wc^T + Wb         (WMMA f32)
//   2. scores_init  : si/sj[N,4] = Wx . a_i / a_j ; init m=-inf, den=0, agg=0
//   3. edge_max     : segment-max of leaky-relu scores (float atomicMax)
//   4. edge_accum   : den[dst] += ex ; agg[dst,:] += ex * Wx[src,:]
//                     (sum(ex*Wx)/den == sum(alpha*Wx), so normalize later)
//   5. norm_colmax  : g = agg/den (in place) + per-column max over node axis
//   6. colsum       : per-column sum of exp(g - colmax)
//   7. gemm_out     : out[N,64] = softmax_col(g) @ Wo_w^T + Wo_b (WMMA f32,
//                     softmax fused into A-fragment load; h_cat never stored)
// ---------------------------------------------------------------------------

typedef __attribute__((ext_vector_type(2))) float v2f;
typedef __attribute__((ext_vector_type(8))) float v8f;

#define DX   128
#define DH   64
#define KH   4
#define CDIM 256   // K*DH
#define NEG_SLOPE 0.01f

// WMMA f32 16x16x4 fragment map (32-bit operands, wave32), from CDNA5 ISA:
//   A (16x4 MxK): lane<16: M=lane, k={k0,k0+1}; lane>=16: M=lane-16, k={k0+2,k0+3}
//   B (4x16 KxN): lane<16: N=lane, same k split -> float2 at Wrowmajor[n*ld + kb]
//   C/D 16x16   : vgpr r: lanes0-15 M=r, lanes16-31 M=r+8, N=lane&15

__global__ __launch_bounds__(256) void gemm_xw_kernel(
    const float* __restrict__ x, const float* __restrict__ Wc,
    const float* __restrict__ Wb, float* __restrict__ Wx, int nNodes)
{
  const int wave = threadIdx.x >> 5;
  const int lane = threadIdx.x & 31;
  const int tile = blockIdx.x * 8 + wave;
  const int mtiles = nNodes >> 4;
  if (tile >= mtiles * 4) return;                 // wave-uniform: EXEC all-ones inside

  const int mt = tile >> 2;
  const int n0 = (tile & 3) << 6;                 // 64-column head group
  const int row = (mt << 4) + (lane & 15);        // A-fragment row
  const int kh  = (lane >> 4) << 1;               // 0 or 2
  const int nn  = lane & 15;                      // B/D column within 16-tile

  v8f a0 = {}, a1 = {}, a2 = {}, a3 = {};
  const float* arow = x + (size_t)row * DX;
  const float* b0p = Wc + (size_t)(n0 + nn)      * DX;
  const float* b1p = Wc + (size_t)(n0 + nn + 16) * DX;
  const float* b2p = Wc + (size_t)(n0 + nn + 32) * DX;
  const float* b3p = Wc + (size_t)(n0 + nn + 48) * DX;

  for (int k = 0; k < DX; k += 4) {
    const int kb = k + kh;
    v2f av = *(const v2f*)(arow + kb);
    v2f b0 = *(const v2f*)(b0p + kb);
    v2f b1 = *(const v2f*)(b1p + kb);
    v2f b2 = *(const v2f*)(b2p + kb);
    v2f b3 = *(const v2f*)(b3p + kb);
    a0 = __builtin_amdgcn_wmma_f32_16x16x4_f32(false, av, false, b0, (short)0, a0, false, false);
    a1 = __builtin_amdgcn_wmma_f32_16x16x4_f32(false, av, false, b1, (short)0, a1, false, false);
    a2 = __builtin_amdgcn_wmma_f32_16x16x4_f32(false, av, false, b2, (short)0, a2, false, false);
    a3 = __builtin_amdgcn_wmma_f32_16x16x4_f32(false, av, false, b3, (short)0, a3, false, false);
  }

  const int mbase = (mt << 4) + ((lane >> 4) << 3);
  #pragma unroll
  for (int r = 0; r < 8; ++r) {
    float* o = Wx + (size_t)(mbase + r) * CDIM + n0;
    o[nn]      = a0[r] + Wb[n0 + nn];
    o[nn + 16] = a1[r] + Wb[n0 + nn + 16];
    o[nn + 32] = a2[r] + Wb[n0 + nn + 32];
    o[nn + 48] = a3[r] + Wb[n0 + nn + 48];
  }
}

__global__ __launch_bounds__(256) void scores_init_kernel(
    const float* __restrict__ Wx, const float* __restrict__ aw,
    float* __restrict__ si, float* __restrict__ sj,
    float* __restrict__ mmax, float* __restrict__ den,
    float* __restrict__ agg, int nNodes)
{
  const int t = blockIdx.x * blockDim.x + threadIdx.x;
  if (t >= nNodes * KH) return;
  const int node = t >> 2, h = t & 3;
  const float* wx = Wx + (size_t)node * CDIM + h * DH;
  const float* ai = aw + h * (2 * DH);            // aw: (K,1,128) flat
  float s0 = 0.f, s1 = 0.f;
  #pragma unroll 8
  for (int d = 0; d < DH; ++d) {
    const float w = wx[d];
    s0 += w * ai[d];
    s1 += w * ai[DH + d];
  }
  si[t] = s0; sj[t] = s1;
  mmax[t] = -INFINITY; den[t] = 0.f;
  float* ag = agg + (size_t)node * CDIM + h * DH;
  #pragma unroll 8
  for (int d = 0; d < DH; ++d) ag[d] = 0.f;
}

__global__ void col_init_kernel(float* __restrict__ colmax, float* __restrict__ colsum)
{
  const int c = threadIdx.x;                      // 256 threads
  colmax[c] = -INFINITY; colsum[c] = 0.f;
}

__global__ __launch_bounds__(256) void edge_max_kernel(
    const int* __restrict__ eidx, const float* __restrict__ si,
    const float* __restrict__ sj, const float* __restrict__ ab,
    float* __restrict__ mmax, int E, int etot)
{
  const int t = blockIdx.x * blockDim.x + threadIdx.x;
  if (t >= etot * KH) return;
  const int e = t >> 2, h = t & 3;
  int src, dst;
  if (e < E) { src = eidx[e]; dst = eidx[E + e]; }
  else       { src = dst = e - E; }               // self-loop
  float v = si[dst * KH + h] + sj[src * KH + h] + ab[h];
  v = (v >= 0.f) ? v : NEG_SLOPE * v;
  atomicMax(&mmax[dst * KH + h], v);              // -> global_atomic_max_num_f32
}

// one 256-thread block per edge; lane = (head h, dim d). Coalesced gather of
// Wx[src], f32 atomic scatter into agg[dst] (both L2-resident).
__global__ __launch_bounds__(256) void edge_accum_kernel(
    const int* __restrict__ eidx, const float* __restrict__ si,
    const float* __restrict__ sj, const float* __restrict__ ab,
    const float* __restrict__ mmax, const float* __restrict__ Wx,
    float* __restrict__ den, float* __restrict__ agg, int E)
{
  const int e = blockIdx.x;
  int src, dst;
  if (e < E) { src = eidx[e]; dst = eidx[E + e]; }
  else       { src = dst = e - E; }
  const int h = threadIdx.x >> 6, d = threadIdx.x & 63;
  float v = si[dst * KH + h] + sj[src * KH + h] + ab[h];
  v = (v >= 0.f) ? v : NEG_SLOPE * v;
  const float ex = __expf(v - mmax[dst * KH + h]);
  if (d == 0) atomicAdd(&den[dst * KH + h], ex);
  const float w = Wx[(size_t)src * CDIM + h * DH + d];
  atomicAdd(&agg[(size_t)dst * CDIM + h * DH + d], ex * w);
}

__global__ __launch_bounds__(256) void norm_colmax_kernel(
    float* __restrict__ agg, const float* __restrict__ den,
    float* __restrict__ colmax, int nNodes)
{
  const int c = threadIdx.x;                      // column owner, coalesced sweep
  const int h = c >> 6;
  float lm = -INFINITY;
  for (int n = blockIdx.x; n < nNodes; n += gridDim.x) {
    const float g = agg[(size_t)n * CDIM + c] / den[n * KH + h];
    agg[(size_t)n * CDIM + c] = g;
    lm = fmaxf(lm, g);
  }
  atomicMax(&colmax[c], lm);
}

__global__ __launch_bounds__(256) void colsum_kernel(
    const float* __restrict__ agg, const float* __restrict__ colmax,
    float* __restrict__ colsum, int nNodes)
{
  const int c = threadIdx.x;
  const float cm = colmax[c];
  float s = 0.f;
  for (int n = blockIdx.x; n < nNodes; n += gridDim.x)
    s += __expf(agg[(size_t)n * CDIM + c] - cm);
  atomicAdd(&colsum[c], s);
}

__global__ __launch_bounds__(256) void gemm_out_kernel(
    const float* __restrict__ g, const float* __restrict__ colmax,
    const float* __restrict__ colsum, const float* __restrict__ Wo_w,
    const float* __restrict__ Wo_b, float* __restrict__ out, int nNodes)
{
  const int wave = threadIdx.x >> 5;
  const int lane = threadIdx.x & 31;
  const int mt = blockIdx.x * 8 + wave;
  const int mtiles = nNodes >> 4;
  if (mt >= mtiles) return;                       // wave-uniform

  const int row = (mt << 4) + (lane & 15);
  const int kh  = (lane >> 4) << 1;
  const int nn  = lane & 15;

  v8f a0 = {}, a1 = {}, a2 = {}, a3 = {};
  const float* arow = g + (size_t)row * CDIM;
  const float* b0p = Wo_w + (size_t)(nn)      * CDIM;
  const float* b1p = Wo_w + (size_t)(nn + 16) * CDIM;
  const float* b2p = Wo_w + (size_t)(nn + 32) * CDIM;
  const float* b3p = Wo_w + (size_t)(nn + 48) * CDIM;

  for (int k = 0; k < CDIM; k += 4) {
    const int kb = k + kh;
    v2f av = *(const v2f*)(arow + kb);
    av[0] = __expf(av[0] - colmax[kb])     / colsum[kb];     // fused column softmax
    av[1] = __expf(av[1] - colmax[kb + 1]) / colsum[kb + 1];
    v2f b0 = *(const v2f*)(b0p + kb);
    v2f b1 = *(const v2f*)(b1p + kb);
    v2f b2 = *(const v2f*)(b2p + kb);
    v2f b3 = *(const v2f*)(b3p + kb);
    a0 = __builtin_amdgcn_wmma_f32_16x16x4_f32(false, av, false, b0, (short)0, a0, false, false);
    a1 = __builtin_amdgcn_wmma_f32_16x16x4_f32(false, av, false, b1, (short)0, a1, false, false);
    a2 = __builtin_amdgcn_wmma_f32_16x16x4_f32(false, av, false, b2, (short)0, a2, false, false);
    a3 = __builtin_amdgcn_wmma_f32_16x16x4_f32(false, av, false, b3, (short)0, a3, false, false);
  }

  const int mbase = (mt << 4) + ((lane >> 4) << 3);
  #pragma unroll
  for (int r = 0; r < 8; ++r) {
    float* o = out + (size_t)(mbase + r) * DH;
    o[nn]      = a0[r] + Wo_b[nn];
    o[nn + 16] = a1[r] + Wo_b[nn + 16];
    o[nn + 32] = a2[r] + Wo_b[nn + 32];
    o[nn + 48] = a3[r] + Wo_b[nn + 48];
  }
}

extern "C" void kernel_launch(void* const* d_in, const int* in_sizes, int n_in,
                              void* d_out, int out_size, void* d_ws, size_t ws_size,
                              hipStream_t stream) {
  const int*   edge_index = (const int*)  d_in[0];   // (2,E) flat: [src..., dst...]
  const float* x          = (const float*)d_in[1];   // (N,128)
  const float* Ww         = (const float*)d_in[2];   // (4,64,128) == Wc[256,128]
  const float* Wb         = (const float*)d_in[3];   // (4,64)     == [256]
  const float* aw         = (const float*)d_in[4];   // (4,1,128)
  const float* ab         = (const float*)d_in[5];   // (4,1)
  const float* Wo_w       = (const float*)d_in[6];   // (64,256)
  const float* Wo_b       = (const float*)d_in[7];   // (64,)
  float* out = (float*)d_out;

  const int E = in_sizes[0] / 2;
  const int N = in_sizes[1] / DX;                    // 50000 = 3125*16
  const int etot = E + N;                            // edges + self-loops

  // Workspace layout (floats): 2*N*256 + 4*N*4 + 512 ≈ 103 MB
  float* Wx     = (float*)d_ws;                      // N*256
  float* agg    = Wx   + (size_t)N * CDIM;           // N*256 (becomes g in place)
  float* si     = agg  + (size_t)N * CDIM;           // N*4
  float* sj     = si   + (size_t)N * KH;             // N*4
  float* mmax   = sj   + (size_t)N * KH;             // N*4
  float* den    = mmax + (size_t)N * KH;             // N*4
  float* colmax = den  + (size_t)N * KH;             // 256
  float* colsum = colmax + CDIM;                     // 256

  const int tiles1 = (N >> 4) * 4;
  gemm_xw_kernel<<<(tiles1 + 7) / 8, 256, 0, stream>>>(x, Ww, Wb, Wx, N);

  const int tsc = N * KH;
  scores_init_kernel<<<(tsc + 255) / 256, 256, 0, stream>>>(Wx, aw, si, sj, mmax, den, agg, N);
  col_init_kernel<<<1, 256, 0, stream>>>(colmax, colsum);

  const int tem = etot * KH;
  edge_max_kernel<<<(tem + 255) / 256, 256, 0, stream>>>(edge_index, si, sj, ab, mmax, E, etot);

  edge_accum_kernel<<<etot, 256, 0, stream>>>(edge_index, si, sj, ab, mmax, Wx, den, agg, E);

  norm_colmax_kernel<<<512, 256, 0, stream>>>(agg, den, colmax, N);
  colsum_kernel<<<512, 256, 0, stream>>>(agg, colmax, colsum, N);

  const int tiles2 = N >> 4;
  gemm_out_kernel<<<(tiles2 + 7) / 8, 256, 0, stream>>>(agg, colmax, colsum, Wo_w, Wo_b, out, N);
}